// HashEmbedder_61821759259412
// MI455X (gfx1250) — compile-verified
//
#include <hip/hip_runtime.h>
#include <math.h>

#define NLV      16
#define TPB      256
#define PIV_LOG  10
#define PIV      (1 << PIV_LOG)   // 1024 pivots per level
#define SENTINEL 0x7fffffff

typedef __attribute__((ext_vector_type(2))) float v2f;

struct Level {
    const int*   tbl;   // sorted hash table (int32)
    const float* emb;   // [size][2] float32
    int          size;
    int          k;     // smallest k with (1<<k) >= size
    int          s;     // k > PIV_LOG ? k - PIV_LOG : 0  (pivot stride shift)
    int          res;
    float        fres;
    int          pad;
};

struct Params {
    const float* x;     // [N][3]
    float*       out;   // [N][32]
    int          N;
    int          pad;
    Level        lv[NLV];
};

__global__ __launch_bounds__(TPB)
void hashenc_kernel(Params p) {
    extern __shared__ int piv[];                 // NLV * PIV ints = 64 KB

    // ---- phase 0: sentinel-fill pivot cache (OOB pivots act as +inf) ----
    for (int slot = threadIdx.x; slot < NLV * PIV; slot += TPB)
        piv[slot] = SENTINEL;
    __syncthreads();

    // ---- phase 1: async scattered gather of pivots, global -> LDS ------
    // pivot m of level l caches tbl[((m+1)<<s) - 1]; per-lane global and
    // per-lane LDS addresses -> one global_load_async_to_lds_b32 per slot.
    for (int slot = threadIdx.x; slot < NLV * PIV; slot += TPB) {
        const int l = slot >> PIV_LOG;
        const int m = slot & (PIV - 1);
        const Level& L = p.lv[l];
        const int np = (m + 1) << L.s;           // <= 2^k, fits int
        if (np <= L.size) {
            unsigned           lds_addr = (unsigned)(size_t)&piv[slot];
            unsigned long long gaddr    = (unsigned long long)(size_t)&L.tbl[np - 1];
            asm volatile("global_load_async_to_lds_b32 %0, %1, off"
                         :: "v"(lds_addr), "v"(gaddr) : "memory");
        }
    }
#if __has_builtin(__builtin_amdgcn_s_wait_asynccnt)
    __builtin_amdgcn_s_wait_asynccnt(0);
#else
    asm volatile("s_wait_asynccnt 0" ::: "memory");
#endif
    __syncthreads();

    // ---- phase 2: per-point encode ------------------------------------
    const int n = blockIdx.x * TPB + threadIdx.x;
    if (n >= p.N) return;

    const float xv = p.x[3 * n + 0];
    const float yv = p.x[3 * n + 1];
    const float zv = p.x[3 * n + 2];
    float* outp = p.out + (size_t)n * (NLV * 2);

    for (int l = 0; l < NLV; ++l) {
        const Level L = p.lv[l];
        const float gridw = 2.0f / L.fres;

        const float blx = floorf((xv + 1.0f) / gridw);
        const float bly = floorf((yv + 1.0f) / gridw);
        const float blz = floorf((zv + 1.0f) / gridw);
        const float wx = (xv - (blx * gridw - 1.0f)) / gridw;
        const float wy = (yv - (bly * gridw - 1.0f)) / gridw;
        const float wz = (zv - (blz * gridw - 1.0f)) / gridw;

        const int res  = L.res;
        const int res2 = res * res;
        const int base = (int)blx + (int)bly * res + (int)blz * res2;

        int key[8];
#pragma unroll
        for (int c = 0; c < 8; ++c)
            key[c] = base + ((c >> 2) & 1) + ((c >> 1) & 1) * res + (c & 1) * res2;

        // lockstep branchless rank search: 8 independent loads in flight
        // per bit iteration; pivot levels (b >= s) come from LDS.
        const int* pv = piv + (l << PIV_LOG);
        int pos[8] = {0, 0, 0, 0, 0, 0, 0, 0};
        for (int b = L.k - 1; b >= 0; --b) {
            const int bit = 1 << b;
            if (b >= L.s) {
#pragma unroll
                for (int c = 0; c < 8; ++c) {
                    const int np = pos[c] + bit;
                    const int v  = pv[(np >> L.s) - 1];      // ds_load
                    if (v < key[c]) pos[c] = np;
                }
            } else {
#pragma unroll
                for (int c = 0; c < 8; ++c) {
                    const int np = pos[c] + bit;
                    const int v  = (np <= L.size) ? L.tbl[np - 1] : SENTINEL;
                    if (v < key[c]) pos[c] = np;
                }
            }
        }

        // gather embeddings + trilinear blend
        float a0 = 0.0f, a1 = 0.0f;
#pragma unroll
        for (int c = 0; c < 8; ++c) {
            const int row = (pos[c] < L.size) ? pos[c] : (L.size - 1);
            const v2f e = *(const v2f*)(L.emb + 2 * (size_t)row);
            const float wfx = (c & 4) ? wx : 1.0f - wx;
            const float wfy = (c & 2) ? wy : 1.0f - wy;
            const float wfz = (c & 1) ? wz : 1.0f - wz;
            const float w = wfx * wfy * wfz;
            a0 += w * e.x;
            a1 += w * e.y;
        }

        // write-once output: non-temporal so it doesn't evict L2-resident tables
        v2f o = {a0, a1};
        __builtin_nontemporal_store(o, (v2f*)(outp + 2 * l));
    }
}

extern "C" void kernel_launch(void* const* d_in, const int* in_sizes, int n_in,
                              void* d_out, int out_size, void* d_ws, size_t ws_size,
                              hipStream_t stream) {
    (void)n_in; (void)d_ws; (void)ws_size; (void)out_size;

    Params p;
    p.x   = (const float*)d_in[0];
    p.out = (float*)d_out;
    p.N   = in_sizes[0] / 3;
    p.pad = 0;

    // replicate RESOLUTIONS = floor(16 * b^i), b = exp((ln 512 - ln 16)/15)
    const double B = exp((log(512.0) - log(16.0)) / 15.0);
    for (int l = 0; l < NLV; ++l) {
        const int res  = (int)floor(16.0 * pow(B, (double)l));
        const int size = in_sizes[1 + l];
        Level& L = p.lv[l];
        L.tbl  = (const int*)d_in[1 + l];
        L.emb  = (const float*)d_in[1 + NLV + l];
        L.size = size;
        int k = 0;
        while ((1ll << k) < (long long)size) ++k;
        L.k    = k;
        L.s    = (k > PIV_LOG) ? (k - PIV_LOG) : 0;
        L.res  = res;
        L.fres = (float)res;
        L.pad  = 0;
    }

    const int blocks = (p.N + TPB - 1) / TPB;
    hashenc_kernel<<<blocks, TPB, NLV * PIV * sizeof(int), stream>>>(p);
}